// RcsFullStitchCrop2_29910152249782
// MI455X (gfx1250) — compile-verified
//
#include <hip/hip_runtime.h>
#include <stdint.h>

// Problem geometry (fixed by the reference: H=W=3072, C=2, 64x64 tiles).
#define KSIZE     64
#define HH        3072
#define WW        3072
#define CC        2
#define TWID      (WW / KSIZE)                    // 48 tiles per row
#define NTILES    ((HH / KSIZE) * (WW / KSIZE))   // 2304 tiles
#define TILE_ELEMS (CC * KSIZE * KSIZE)           // 8192 floats = 32 KB
#define HWSZ      (HH * WW)
#define F4_PER_TILE (TILE_ELEMS / 4)              // 2048 float4 chunks
#define BLOCK     256
#define CHUNKS    (F4_PER_TILE / BLOCK)           // 8 chunks / thread

// CDNA5 async LDS DMA path (gfx1250): global -> LDS -> global, tracked by ASYNCcnt.
#if defined(__HIP_DEVICE_COMPILE__) && defined(__gfx1250__) && \
    __has_builtin(__builtin_amdgcn_global_load_async_to_lds_b128) && \
    __has_builtin(__builtin_amdgcn_global_store_async_from_lds_b128)
#define USE_ASYNC_LDS 1
#else
#define USE_ASYNC_LDS 0
#endif

// The builtin expects pointers to a 16-byte int vector (per hipcc diagnostic):
//   param0: v4i in the global (addrspace(1)) space
//   param1: v4i in the LDS    (addrspace(3)) space
typedef int v4i __attribute__((ext_vector_type(4)));
typedef __attribute__((address_space(1))) v4i gv4i_t;
typedef __attribute__((address_space(3))) v4i lv4i_t;

__device__ __forceinline__ gv4i_t* to_global(const void* p) {
  // Global generic pointers carry the full 64-bit address.
  return (gv4i_t*)(uintptr_t)p;
}
__device__ __forceinline__ lv4i_t* to_lds(const void* p) {
  // Generic LDS address = {SHARED_BASE, offset[31:0]}; low 32 bits are the LDS offset.
  return (lv4i_t*)(uint32_t)(uintptr_t)p;
}

// Scatter: block j copies input tile j (strided [C,64,64] view of [C,H,W])
// to contiguous output tile perm[j]. blockIdx.y selects the x / y array.
__global__ __launch_bounds__(BLOCK) void tile_scatter_kernel(
    const float* __restrict__ x, const float* __restrict__ y,
    const int* __restrict__ perm, float* __restrict__ out) {
  const int j   = blockIdx.x;
  const int a   = blockIdx.y;
  const int tid = threadIdx.x;

  const float* src = (a == 0) ? x : y;
  const int dstTile = perm[j];
  float* dst = out + (size_t)a * (size_t)NTILES * TILE_ELEMS
                   + (size_t)dstTile * TILE_ELEMS;

  const int ph = j / TWID;
  const int pw = j - ph * TWID;
  const float* tbase = src + (size_t)(ph * KSIZE) * WW + (size_t)(pw * KSIZE);

#if USE_ASYNC_LDS
  __shared__ float smem[TILE_ELEMS];  // 32 KB staging tile

  // Phase 1: strided global tile -> LDS (async, 16B per lane per op).
#pragma unroll
  for (int k = 0; k < CHUNKS; ++k) {
    const int f   = k * BLOCK + tid;   // float4 index in [0, 2048)
    const int c   = f >> 10;           // channel (1024 float4 per channel)
    const int rem = f & 1023;
    const int r   = rem >> 4;          // row in tile (16 float4 per row)
    const int c4  = rem & 15;          // float4 column
    const float* g = tbase + (size_t)c * HWSZ + (size_t)r * WW + (size_t)(c4 * 4);
    __builtin_amdgcn_global_load_async_to_lds_b128(
        to_global(g), to_lds(&smem[f * 4]), /*offset=*/0, /*cpol=*/0);
  }
  // Each thread stores exactly the LDS bytes it loaded -> only need ASYNCcnt==0,
  // no workgroup barrier.
  asm volatile("s_wait_asynccnt 0" ::: "memory");

  // Phase 2: LDS -> contiguous permuted output tile (async).
#pragma unroll
  for (int k = 0; k < CHUNKS; ++k) {
    const int f = k * BLOCK + tid;
    __builtin_amdgcn_global_store_async_from_lds_b128(
        to_global(dst + f * 4), to_lds(&smem[f * 4]), /*offset=*/0, /*cpol=*/0);
  }
  asm volatile("s_wait_asynccnt 0" ::: "memory");
#else
  // Fallback: direct b128 copy through VGPRs (still fully coalesced).
#pragma unroll
  for (int k = 0; k < CHUNKS; ++k) {
    const int f   = k * BLOCK + tid;
    const int c   = f >> 10;
    const int rem = f & 1023;
    const int r   = rem >> 4;
    const int c4  = rem & 15;
    const float4 v = *reinterpret_cast<const float4*>(
        tbase + (size_t)c * HWSZ + (size_t)r * WW + (size_t)(c4 * 4));
    *reinterpret_cast<float4*>(dst + f * 4) = v;
  }
#endif
}

extern "C" void kernel_launch(void* const* d_in, const int* in_sizes, int n_in,
                              void* d_out, int out_size, void* d_ws, size_t ws_size,
                              hipStream_t stream) {
  const float* x    = (const float*)d_in[0];
  const float* y    = (const float*)d_in[1];
  const int*   perm = (const int*)d_in[2];
  float*       out  = (float*)d_out;

  dim3 grid(NTILES, 2, 1);
  tile_scatter_kernel<<<grid, dim3(BLOCK, 1, 1), 0, stream>>>(x, y, perm, out);

  (void)in_sizes; (void)n_in; (void)out_size; (void)d_ws; (void)ws_size;
}